// FastSamPredict_62448824484519
// MI455X (gfx1250) — compile-verified
//
#include <hip/hip_runtime.h>
#include <hip/hip_bf16.h>
#include <limits.h>

// Problem constants (from reference)
#define MASKS   64
#define H       1024
#define W       1024
#define SEGS    32               // blocks per mask; each block = 32 full rows
#define TILES   8                // 16 KB tiles per block (4 rows each)
#define TILE_F  4096             // floats per tile (16 KB)
#define NBUF    4                // quad-buffered LDS staging
#define THREADS 256              // 8 wave32s

typedef unsigned int uint32x4 __attribute__((ext_vector_type(4)));
typedef int          int32x8  __attribute__((ext_vector_type(8)));
typedef int          int32x4  __attribute__((ext_vector_type(4)));

// Issue one TDM descriptor: 1-D copy of 16 KB (2048 x 8-byte elements)
// from global address `gaddr` into LDS byte offset `lds_off`.
// D# layout per CDNA5 ISA ch.8:
//   group0: [1:0]=count=1, [63:32]=lds_addr, [120:64]=global_addr, [127:126]=type=2
//   group1: [17:16]=data_size=3 (8B), tensor_dim0=2048 (bits 79:48),
//           tensor_dim1=1 (bits 111:80), tile_dim0=2048 (bits 127:112),
//           tile_dim1/2=0 (unused), tensor_dim0_stride=2048 (bits 207:160)
__device__ __forceinline__ void tdm_load_16k(unsigned long long gaddr,
                                             unsigned int lds_off) {
    uint32x4 g0;
    g0.x = 1u;                                                   // count=1, user mode
    g0.y = lds_off;                                              // LDS byte address
    g0.z = (unsigned int)(gaddr & 0xFFFFFFFFull);                // global_addr[31:0]
    g0.w = (unsigned int)((gaddr >> 32) & 0x1FFFFFFull)          // global_addr[56:32]
         | (2u << 30);                                           // type = 2 ("image")
    int32x8 g1;
    g1[0] = 0x00030000;   // workgroup_mask=0 (no cluster), data_size=3 (8 bytes)
    g1[1] = 0x08000000;   // tensor_dim0[15:0]=0x800 in bits 63:48
    g1[2] = 0x00010000;   // tensor_dim0[31:16]=0, tensor_dim1[15:0]=1
    g1[3] = 0x08000000;   // tensor_dim1[31:16]=0, tile_dim0=0x800 (2048 elems)
    g1[4] = 0;            // tile_dim1=0, tile_dim2=0 (1-D tile)
    g1[5] = 2048;         // tensor_dim0_stride[31:0]
    g1[6] = 0;
    g1[7] = 0;
    int32x4 z4 = {0, 0, 0, 0};
#if defined(__clang_major__) && (__clang_major__ >= 23)
    int32x8 z8 = {0, 0, 0, 0, 0, 0, 0, 0};
    __builtin_amdgcn_tensor_load_to_lds(g0, g1, z4, z4, z8, 0);
#else
    __builtin_amdgcn_tensor_load_to_lds(g0, g1, z4, z4, 0);
#endif
}

// Re-seed per-mask reduction state every call (harness does not re-poison ws).
// ws layout per mask: [min_col, min_row, max_col, max_row, count]
__global__ void bbox_init(int* __restrict__ ws) {
    int m = threadIdx.x;
    if (m < MASKS) {
        int* w = ws + m * 5;
        w[0] = INT_MAX; w[1] = INT_MAX; w[2] = -1; w[3] = -1; w[4] = 0;
    }
}

__global__ __launch_bounds__(THREADS)
void bbox_main(const float* __restrict__ masks, int* __restrict__ ws) {
    __shared__ __align__(16) float tile_buf[NBUF][TILE_F];   // 64 KB quad buffer
    __shared__ int s_minc[8], s_minr[8], s_maxc[8], s_maxr[8], s_cnt[8];

    const int mask = blockIdx.y;
    const int seg  = blockIdx.x;
    const int tid  = threadIdx.x;
    const int wave = tid >> 5;

    // byte address of this block's 128 KB segment (rows seg*32 .. seg*32+31)
    const unsigned long long base =
        (unsigned long long)(size_t)masks
        + ((unsigned long long)mask << 22)    // mask * 1M pixels * 4 B
        + ((unsigned long long)seg  << 17);   // seg  * 32K pixels * 4 B

    unsigned int lds_off[NBUF];
#pragma unroll
    for (int b = 0; b < NBUF; ++b)
        lds_off[b] = (unsigned int)(size_t)&tile_buf[b][0];

    // Prime the pipeline: 3 tiles in flight before consuming tile 0.
    if (wave == 0) {
        tdm_load_16k(base,                               lds_off[0]);
        tdm_load_16k(base + (1ull << 14),                lds_off[1]);
        tdm_load_16k(base + (2ull << 14),                lds_off[2]);
    }

    // Per-thread state. Key index algebra: every stride is a multiple of W,
    // so this thread always reads columns tid*4 .. tid*4+3, and iteration
    // (t,k) is exactly row seg*32 + t*4 + k (same for all threads).
    unsigned int c0 = 0, c1 = 0, c2 = 0, c3 = 0;  // per-column "any" (bit OR)
    unsigned int rowbits = 0;                     // bit i -> row seg*32+i active
    float cntf = 0.0f;                            // active-pixel count (exact)

#pragma unroll
    for (int t = 0; t < TILES; ++t) {
        if (wave == 0) {
            if (t < TILES - 3) {
                tdm_load_16k(base + ((unsigned long long)(t + 3) << 14),
                             lds_off[(t + 3) & (NBUF - 1)]);
                __builtin_amdgcn_s_wait_tensorcnt(3);  // tile t done, 3 in flight
            } else if (t == TILES - 3) {
                __builtin_amdgcn_s_wait_tensorcnt(2);
            } else if (t == TILES - 2) {
                __builtin_amdgcn_s_wait_tensorcnt(1);
            } else {
                __builtin_amdgcn_s_wait_tensorcnt(0);
            }
        }
        __syncthreads();                        // DMA'd tile visible to all waves

        const float4* src = (const float4*)&tile_buf[t & (NBUF - 1)][0];
#pragma unroll
        for (int k = 0; k < 4; ++k) {
            float4 f = src[k * THREADS + tid];                 // ds_load_b128
            unsigned int bx = __float_as_uint(f.x);
            unsigned int by = __float_as_uint(f.y);
            unsigned int bz = __float_as_uint(f.z);
            unsigned int bw = __float_as_uint(f.w);
            c0 |= bx; c1 |= by; c2 |= bz; c3 |= bw;
            unsigned int u = (bx | by) | (bz | bw);
            rowbits |= (u != 0u) ? (1u << (t * 4 + k)) : 0u;
            cntf += (f.x + f.y) + (f.z + f.w);                 // binary -> count
        }
        __syncthreads();                        // all reads done before reuse
    }

    // Recover per-thread bbox extents once (instead of per pixel).
    const int cb = tid * 4;   // tid < 256 -> no mod-1024 wrap
    unsigned int colmask = (c0 ? 1u : 0u) | (c1 ? 2u : 0u)
                         | (c2 ? 4u : 0u) | (c3 ? 8u : 0u);
    int minc, maxc, minr, maxr;
    if (colmask) {
        minc = cb + (__ffs(colmask) - 1);
        maxc = cb + (31 - __clz(colmask));
    } else { minc = INT_MAX; maxc = -1; }
    if (rowbits) {
        minr = seg * 32 + (__ffs(rowbits) - 1);
        maxr = seg * 32 + (31 - __clz(rowbits));
    } else { minr = INT_MAX; maxr = -1; }
    int cnt = (int)cntf;

    // wave32 shuffle reduction
    for (int off = 16; off > 0; off >>= 1) {
        minc = min(minc, __shfl_down(minc, off, 32));
        minr = min(minr, __shfl_down(minr, off, 32));
        maxc = max(maxc, __shfl_down(maxc, off, 32));
        maxr = max(maxr, __shfl_down(maxr, off, 32));
        cnt += __shfl_down(cnt, off, 32);
    }
    if ((tid & 31) == 0) {
        s_minc[wave] = minc; s_minr[wave] = minr;
        s_maxc[wave] = maxc; s_maxr[wave] = maxr; s_cnt[wave] = cnt;
    }
    __syncthreads();
    if (tid == 0) {
        for (int i = 1; i < 8; ++i) {
            minc = min(minc, s_minc[i]); minr = min(minr, s_minr[i]);
            maxc = max(maxc, s_maxc[i]); maxr = max(maxr, s_maxr[i]);
            cnt += s_cnt[i];
        }
        int* w = ws + mask * 5;
        atomicMin(&w[0], minc);
        atomicMin(&w[1], minr);
        atomicMax(&w[2], maxc);
        atomicMax(&w[3], maxr);
        atomicAdd(&w[4], cnt);
    }
}

// area_thresh = (1024/32)*(1024/32) = 1024 ; keep iff sum > 1024 (binary mask)
__global__ void bbox_finalize(const int* __restrict__ ws, float* __restrict__ out) {
    int m = threadIdx.x;
    if (m < MASKS) {
        const int* w = ws + m * 5;
        bool keep = w[4] > 1024;
        out[m * 4 + 0] = keep ? (float)w[0] : 0.0f;   // x1 = min col
        out[m * 4 + 1] = keep ? (float)w[1] : 0.0f;   // y1 = min row
        out[m * 4 + 2] = keep ? (float)w[2] : 0.0f;   // x2 = max col
        out[m * 4 + 3] = keep ? (float)w[3] : 0.0f;   // y2 = max row
    }
}

extern "C" void kernel_launch(void* const* d_in, const int* in_sizes, int n_in,
                              void* d_out, int out_size, void* d_ws, size_t ws_size,
                              hipStream_t stream) {
    const float* masks = (const float*)d_in[0];   // [64,1024,1024] fp32 in {0,1}
    int*   ws  = (int*)d_ws;                      // 64*5 ints of reduction state
    float* out = (float*)d_out;                   // [64,4] fp32

    bbox_init<<<1, 64, 0, stream>>>(ws);
    dim3 grid(SEGS, MASKS);                       // 32 x 64 = 2048 blocks
    bbox_main<<<grid, THREADS, 0, stream>>>(masks, ws);
    bbox_finalize<<<1, 64, 0, stream>>>(ws, out);
}